// Gemma4VisionPatchEmbedder_2035814498742
// MI455X (gfx1250) — compile-verified
//
#include <hip/hip_runtime.h>
#include <cstdint>

// ---- problem constants (match reference) ----
#define HIDDEN   1152
#define IN_DIM   768
#define NROWS    (16 * 4096)   // B * NP
#define POS_SIZE 64

// ---- tiling ----
#define BM   128
#define BN   128
#define BK   32
#define KPAD 4
#define LDK  (BK + KPAD)          // 36 floats -> 144B row stride, 16B aligned
#define NCHUNK (IN_DIM / BK)      // 24

typedef __attribute__((ext_vector_type(2))) float v2f;
typedef __attribute__((ext_vector_type(8))) float v8f;

__global__ __launch_bounds__(256)
void patch_embed_wmma_kernel(const float* __restrict__ pixel,          // [NROWS, IN_DIM]
                             const long long* __restrict__ pos,        // [NROWS, 2] int64
                             const unsigned char* __restrict__ padmask,// [NROWS] bool
                             const float* __restrict__ W,              // [HIDDEN, IN_DIM]
                             const float* __restrict__ table,          // [2, POS_SIZE, HIDDEN]
                             float* __restrict__ out)                  // [NROWS, HIDDEN]
{
    __shared__ float As[BM * LDK];
    __shared__ float Bs[BN * LDK];
    __shared__ int   sPosX[BM];
    __shared__ int   sPosY[BM];
    __shared__ int   sPad[BM];

    const int tid  = threadIdx.x;     // 0..255 (8 waves, wave32)
    const int wave = tid >> 5;        // 0..7
    const int lane = tid & 31;
    const int lo16 = lane & 15;
    const int hi   = lane >> 4;       // 0: K=0/1 half, 1: K=2/3 half

    const int mTile = blockIdx.y * BM;   // output-row base
    const int hTile = blockIdx.x * BN;   // output-col base

    // Per-row positional metadata -> LDS (one int64 gather per row, done once)
    if (tid < BM) {
        const int r = mTile + tid;
        long long px = pos[2LL * r + 0];
        long long py = pos[2LL * r + 1];
        sPosX[tid] = (int)(px < 0 ? 0 : px);
        sPosY[tid] = (int)(py < 0 ? 0 : py);
        sPad[tid]  = (int)padmask[r];
    }

    v8f acc[8] = {};   // 16x128 fp32 strip per wave

    // Loader mapping: thread t moves one 16-float half row of A and of W-tile
    const int lm   = tid >> 1;          // 0..127
    const int lseg = (tid & 1) * 16;    // 0 or 16
    const float* aRow = pixel + (size_t)(mTile + lm) * IN_DIM + lseg;
    const float* bRow = W     + (size_t)(hTile + lm) * IN_DIM + lseg;
    float4* asDst = (float4*)&As[lm * LDK + lseg];
    float4* bsDst = (float4*)&Bs[lm * LDK + lseg];

    for (int kc = 0; kc < NCHUNK; ++kc) {
        const int k0 = kc * BK;
        __syncthreads();

        // global -> LDS (4x float4 each stream; W rows are contiguous in d=k,
        // so the [n][k] LDS layout needs no transpose)
        const float4* ag = (const float4*)(aRow + k0);
        const float4* bg = (const float4*)(bRow + k0);
        #pragma unroll
        for (int i = 0; i < 4; ++i) asDst[i] = ag[i];
        #pragma unroll
        for (int i = 0; i < 4; ++i) bsDst[i] = bg[i];

        // warm caches for the next chunk (global_prefetch_b8)
        if (kc + 1 < NCHUNK) {
            __builtin_prefetch(aRow + k0 + BK, 0, 0);
            __builtin_prefetch(bRow + k0 + BK, 0, 0);
        }
        __syncthreads();

        // A frag (16x4 f32): lane<16 -> (M=lane, K=k..k+1), lane>=16 -> K=k+2..k+3
        const float* aBase = &As[(wave * 16 + lo16) * LDK + hi * 2];
        // B frag (4x16 f32): lane<16 -> (K=k..k+1, N=lane), lane>=16 -> K=k+2..k+3
        const float* bBase = &Bs[lo16 * LDK + hi * 2];

        #pragma unroll
        for (int kk = 0; kk < BK; kk += 4) {
            const v2f afrag = *(const v2f*)(aBase + kk);
            #pragma unroll
            for (int j = 0; j < 8; ++j) {
                const v2f bfrag = *(const v2f*)(bBase + j * 16 * LDK + kk);
                acc[j] = __builtin_amdgcn_wmma_f32_16x16x4_f32(
                    /*neg_a=*/false, afrag,
                    /*neg_b=*/false, bfrag,
                    /*c_mod=*/(short)0, acc[j],
                    /*reuse_a=*/false, /*reuse_b=*/false);
            }
        }
    }

    // Epilogue: C/D layout -> element (VGPR v, lane) = row (v + 8*hi), col lo16.
    const float* t0 = table;                          // table[0]
    const float* t1 = table + (size_t)POS_SIZE * HIDDEN; // table[1]
    #pragma unroll
    for (int j = 0; j < 8; ++j) {
        const int col = hTile + j * 16 + lo16;
        #pragma unroll
        for (int v = 0; v < 8; ++v) {
            const int rl = wave * 16 + v + hi * 8;    // row within block tile
            const int rg = mTile + rl;                // global row
            float pe = 0.f;
            if (!sPad[rl]) {
                pe = t0[(size_t)sPosX[rl] * HIDDEN + col]
                   + t1[(size_t)sPosY[rl] * HIDDEN + col];
            }
            out[(size_t)rg * HIDDEN + col] = acc[j][v] + pe;
        }
    }
}

extern "C" void kernel_launch(void* const* d_in, const int* in_sizes, int n_in,
                              void* d_out, int out_size, void* d_ws, size_t ws_size,
                              hipStream_t stream) {
    (void)in_sizes; (void)n_in; (void)out_size; (void)d_ws; (void)ws_size;
    const float*         pixel = (const float*)d_in[0];
    const long long*     pos   = (const long long*)d_in[1];     // int64 position ids
    const unsigned char* pad   = (const unsigned char*)d_in[2]; // bool mask
    const float*         W     = (const float*)d_in[3];
    const float*         table = (const float*)d_in[4];
    float*               out   = (float*)d_out;

    dim3 grid(HIDDEN / BN, NROWS / BM);   // (9, 512): H-tiles adjacent -> A-tile L2 reuse
    dim3 block(256);                      // 8 wave32 waves
    patch_embed_wmma_kernel<<<grid, block, 0, stream>>>(pixel, pos, pad, W, table, out);
}